// SRNN_43817256354254
// MI455X (gfx1250) — compile-verified
//
#include <hip/hip_runtime.h>
#include <math.h>

typedef __attribute__((ext_vector_type(2))) float v2f;
typedef __attribute__((ext_vector_type(8))) float v8f;

#define N_B   128
#define N_T   512
#define N_IN  256
#define N_REC 1024
#define N_OUT 20
#define N_STEPS 511            // N_T - 1 scan steps

#define ALPHA 0.951229424500714f
#define KAPPA 0.951229424500714f
#define THR   0.6f
#define EPS   1e-5f

// ---------------------------------------------------------------------------
// Init: wr with zeroed diagonal, zero state buffers
// ---------------------------------------------------------------------------
__global__ void srnn_init(const float* __restrict__ w_rec, float* __restrict__ wrD,
                          float* __restrict__ v, float* __restrict__ z0,
                          float* __restrict__ z1) {
  int i = blockIdx.x * blockDim.x + threadIdx.x;
  if (i < N_REC * N_REC) {
    int r = i >> 10, c = i & (N_REC - 1);
    wrD[i] = (r == c) ? 0.0f : w_rec[i];
  }
  if (i < N_B * N_REC) {
    v[i] = 0.0f; z0[i] = 0.0f; z1[i] = 0.0f;
  }
}

// ---------------------------------------------------------------------------
// Precompute input-BN scale/shift per (t, feature): xh = x*sc + sh
// ---------------------------------------------------------------------------
__global__ void srnn_bnin(const float* __restrict__ x, const float* __restrict__ g_in,
                          const float* __restrict__ b_in, float* __restrict__ sc,
                          float* __restrict__ sh) {
  int idx = blockIdx.x * blockDim.x + threadIdx.x;   // t*N_IN + i
  if (idx >= N_STEPS * N_IN) return;
  int t = idx / N_IN, i = idx - t * N_IN;
  const float* xp = x + (size_t)t * N_IN + i;
  float m = 0.0f, sq = 0.0f;
  for (int b = 0; b < N_B; ++b) {
    float xv = xp[(size_t)b * N_T * N_IN];
    m += xv; sq += xv * xv;
  }
  m *= (1.0f / N_B);
  float var = sq * (1.0f / N_B) - m * m;
  float scale = rsqrtf(var + EPS) * g_in[i];
  sc[idx] = scale;
  sh[idx] = b_in[i] - m * scale;
}

// ---------------------------------------------------------------------------
// One recurrent step. Blocks 0..63: each computes the full-batch [128 x 16]
// tile of v1 via f32 WMMA, does intra-WG BatchNorm + threshold, writes new
// v and z (double buffered). Block 64: projects previous step's spikes
// through w_out into s[t-1] (deterministic, no atomics).
// ---------------------------------------------------------------------------
__global__ void __launch_bounds__(256)
srnn_step(int t, const float* __restrict__ x, const float* __restrict__ wrD,
          const float* __restrict__ w_in, const float* __restrict__ w_out,
          const float* __restrict__ g_rec, const float* __restrict__ b_rec,
          const float* __restrict__ sc, const float* __restrict__ sh,
          float* __restrict__ v, const float* __restrict__ zPrev,
          float* __restrict__ zNext, float* __restrict__ s) {
  int tid = threadIdx.x;

  if (blockIdx.x == 64) {
    // s[t-1] = z_{t-1} @ w_out.T  (zPrev is complete: written by step t-1)
    if (t == 0) return;
    for (int p = tid; p < N_B * N_OUT; p += 256) {
      int b = p / N_OUT, o = p - b * N_OUT;
      const float* zr = zPrev + b * N_REC;
      const float* wo = w_out + o * N_REC;
      float acc = 0.0f;
      #pragma unroll 8
      for (int j = 0; j < N_REC; ++j) acc += zr[j] * wo[j];
      s[(size_t)(t - 1) * (N_B * N_OUT) + p] = acc;
    }
    return;
  }

  __shared__ float lds_wr[16 * 256];   // 16 KB: staged wr chunk  [16 feat x 256 k]
  __shared__ float lds_wi[16 * 256];   // 16 KB: staged w_in tile [16 feat x 256 k]
  __shared__ float lds_sc[N_IN];
  __shared__ float lds_sh[N_IN];
  __shared__ float red_sum[8][32];
  __shared__ float red_sq[8][32];
  __shared__ float bn_mean[16];
  __shared__ float bn_rstd[16];

  int wave = tid >> 5, lane = tid & 31;
  int halfsel = lane >> 4;             // which K-half this lane holds
  int lane16 = lane & 15;
  int jBase = blockIdx.x * 16;         // 16 output features per WG
  int mRow = wave * 16 + lane16;       // A-fragment row (batch index)

  // stage w_in tile and input-BN scale/shift for this t
  for (int i = tid; i < 16 * N_IN; i += 256) {
    int r = i >> 8, c = i & 255;
    lds_wi[i] = w_in[(jBase + r) * N_IN + c];
  }
  if (tid < N_IN) {
    lds_sc[tid] = sc[t * N_IN + tid];
    lds_sh[tid] = sh[t * N_IN + tid];
  }

  v8f acc = {};

  // ---- GEMM 1: z_{t-1} @ wrD.T  (K = 1024, staged through LDS in 256-chunks)
  const float* zrow = zPrev + mRow * N_REC;
  for (int kc = 0; kc < N_REC; kc += 256) {
    __syncthreads();
    for (int i = tid; i < 16 * 256; i += 256) {
      int r = i >> 8, c = i & 255;
      lds_wr[i] = wrD[(jBase + r) * N_REC + kc + c];
    }
    __syncthreads();
    #pragma unroll 4
    for (int k = 0; k < 256; k += 4) {
      int ka = k + 2 * halfsel;
      v2f a;  a.x = zrow[kc + ka];            a.y = zrow[kc + ka + 1];
      v2f bb; bb.x = lds_wr[lane16 * 256 + ka]; bb.y = lds_wr[lane16 * 256 + ka + 1];
      acc = __builtin_amdgcn_wmma_f32_16x16x4_f32(false, a, false, bb,
                                                  (short)0, acc, false, false);
    }
  }
  __syncthreads();

  // ---- GEMM 2: BN(x_t) @ w_in.T  (K = 256, BN fused as x*sc+sh)
  const float* xrow = x + (size_t)mRow * N_T * N_IN + (size_t)t * N_IN;
  #pragma unroll 4
  for (int k = 0; k < N_IN; k += 4) {
    int ka = k + 2 * halfsel;
    v2f a;
    a.x = xrow[ka]     * lds_sc[ka]     + lds_sh[ka];
    a.y = xrow[ka + 1] * lds_sc[ka + 1] + lds_sh[ka + 1];
    v2f bb; bb.x = lds_wi[lane16 * N_IN + ka]; bb.y = lds_wi[lane16 * N_IN + ka + 1];
    acc = __builtin_amdgcn_wmma_f32_16x16x4_f32(false, a, false, bb,
                                                (short)0, acc, false, false);
  }

  // ---- epilogue: v1_pre = alpha*v + gemm - thr*z ; intra-WG BatchNorm
  int j = jBase + lane16;
  float vals[8];
  float lsum = 0.0f, lsq = 0.0f;
  #pragma unroll
  for (int r = 0; r < 8; ++r) {
    int m = wave * 16 + r + 8 * halfsel;       // C layout: M = r + 8*(lane>=16)
    float zold = zPrev[m * N_REC + j];
    float vold = v[m * N_REC + j];
    float u = ALPHA * vold + acc[r] - THR * zold;
    vals[r] = u;
    lsum += u; lsq += u * u;
  }
  red_sum[wave][lane] = lsum;
  red_sq[wave][lane]  = lsq;
  __syncthreads();
  if (tid < 16) {
    float sm = 0.0f, sq2 = 0.0f;
    #pragma unroll
    for (int w = 0; w < 8; ++w) {
      sm  += red_sum[w][tid] + red_sum[w][tid + 16];
      sq2 += red_sq[w][tid]  + red_sq[w][tid + 16];
    }
    float mean = sm * (1.0f / N_B);
    float var  = sq2 * (1.0f / N_B) - mean * mean;
    bn_mean[tid] = mean;
    bn_rstd[tid] = rsqrtf(var + EPS);
  }
  __syncthreads();
  float mean = bn_mean[lane16];
  float rstd = bn_rstd[lane16];
  float gam = g_rec[j], bet = b_rec[j];
  #pragma unroll
  for (int r = 0; r < 8; ++r) {
    int m = wave * 16 + r + 8 * halfsel;
    float vn = (vals[r] - mean) * rstd * gam + bet;
    float zn = (vn > THR) ? 1.0f : 0.0f;
    v[m * N_REC + j] = vn;       // carried state is post-BN v1
    zNext[m * N_REC + j] = zn;
  }
}

// ---------------------------------------------------------------------------
// Temporal scan: vo[t] = kappa*vo[t-1] + s[t-1], vo[0] = 0.
// Also computes the last projection s[510] from the final spike buffer.
// ---------------------------------------------------------------------------
__global__ void srnn_scan(const float* __restrict__ s, const float* __restrict__ zFinal,
                          const float* __restrict__ w_out, float* __restrict__ out_vo) {
  int p = blockIdx.x * blockDim.x + threadIdx.x;   // b*20 + o
  if (p >= N_B * N_OUT) return;
  int b = p / N_OUT, o = p - b * N_OUT;
  out_vo[p] = 0.0f;                                // vo[0] = 0
  float a = 0.0f;
  for (int tau = 0; tau < N_STEPS - 1; ++tau) {    // s[0..509] from step kernels
    a = KAPPA * a + s[(size_t)tau * (N_B * N_OUT) + p];
    out_vo[(size_t)(tau + 1) * (N_B * N_OUT) + p] = a;
  }
  const float* zr = zFinal + b * N_REC;
  const float* wo = w_out + o * N_REC;
  float sf = 0.0f;
  #pragma unroll 8
  for (int jj = 0; jj < N_REC; ++jj) sf += zr[jj] * wo[jj];
  a = KAPPA * a + sf;
  out_vo[(size_t)N_STEPS * (N_B * N_OUT) + p] = a;
}

// ---------------------------------------------------------------------------
// softmax over the 20 outputs: [n_t,n_b,20] -> [n_b,n_t,20]
// ---------------------------------------------------------------------------
__global__ void srnn_softmax(const float* __restrict__ out_vo, float* __restrict__ out_sm) {
  int p = blockIdx.x * blockDim.x + threadIdx.x;   // b*N_T + t
  if (p >= N_B * N_T) return;
  int b = p >> 9, t = p & (N_T - 1);
  const float* vp = out_vo + (size_t)t * (N_B * N_OUT) + b * N_OUT;
  float mx = vp[0];
  #pragma unroll
  for (int o = 1; o < N_OUT; ++o) mx = fmaxf(mx, vp[o]);
  float e[N_OUT]; float sum = 0.0f;
  #pragma unroll
  for (int o = 0; o < N_OUT; ++o) { e[o] = __expf(vp[o] - mx); sum += e[o]; }
  float inv = 1.0f / sum;
  float* op = out_sm + (size_t)b * (N_T * N_OUT) + t * N_OUT;
  #pragma unroll
  for (int o = 0; o < N_OUT; ++o) op[o] = e[o] * inv;
}

// ---------------------------------------------------------------------------
extern "C" void kernel_launch(void* const* d_in, const int* in_sizes, int n_in,
                              void* d_out, int out_size, void* d_ws, size_t ws_size,
                              hipStream_t stream) {
  const float* x     = (const float*)d_in[0];
  const float* w_in  = (const float*)d_in[2];
  const float* w_rec = (const float*)d_in[3];
  const float* w_out = (const float*)d_in[4];
  const float* g_in  = (const float*)d_in[5];
  const float* b_in  = (const float*)d_in[6];
  const float* g_rec = (const float*)d_in[7];
  const float* b_rec = (const float*)d_in[8];
  (void)in_sizes; (void)n_in; (void)out_size; (void)ws_size;

  float* ws  = (float*)d_ws;
  float* wrD = ws;                       // 1,048,576
  float* v   = wrD + N_REC * N_REC;      // 131,072
  float* zA  = v   + N_B * N_REC;        // 131,072
  float* zB  = zA  + N_B * N_REC;        // 131,072
  float* sc  = zB  + N_B * N_REC;        // 130,816
  float* sh  = sc  + N_STEPS * N_IN;     // 130,816
  float* s   = sh  + N_STEPS * N_IN;     // 510 * 2560

  float* out_sm = (float*)d_out;                       // [128,512,20]
  float* out_vo = out_sm + (size_t)N_B * N_T * N_OUT;  // [512,128,20]

  srnn_init<<<(N_REC * N_REC + 255) / 256, 256, 0, stream>>>(w_rec, wrD, v, zA, zB);
  srnn_bnin<<<(N_STEPS * N_IN + 255) / 256, 256, 0, stream>>>(x, g_in, b_in, sc, sh);

  for (int t = 0; t < N_STEPS; ++t) {
    float* zp = (t & 1) ? zB : zA;
    float* zn = (t & 1) ? zA : zB;
    srnn_step<<<65, 256, 0, stream>>>(t, x, wrD, w_in, w_out, g_rec, b_rec,
                                      sc, sh, v, zp, zn, s);
  }
  // final z buffer: step 510 (even) wrote zB
  srnn_scan<<<(N_B * N_OUT + 255) / 256, 256, 0, stream>>>(s, zB, w_out, out_vo);
  srnn_softmax<<<(N_B * N_T + 255) / 256, 256, 0, stream>>>(out_vo, out_sm);
}